// PILossOBJ_60756607369415
// MI455X (gfx1250) — compile-verified
//
#include <hip/hip_runtime.h>
#include <math.h>

typedef __attribute__((ext_vector_type(2))) float v2f;
typedef __attribute__((ext_vector_type(8))) float v8f;

#define N_VIEWS 12
#define BATCH   4096
#define DIM     128
#define ALPHA_C 0.4f
#define BETA_C  2.0f
#define LAMDA_C 2.0f

// ws layout (6 dwords): [0]=sumLoss [1]=count [2]=sum sqrt(mc) [3]=sum sqrt(a_clstr)
//                       [4]=sum sqrt(c_intra) [5]=min bits (uint, +inf init)

__global__ void pi_init(float* ws) {
    int t = threadIdx.x;
    if (t < 5) ws[t] = 0.0f;
    if (t == 5) ((unsigned*)ws)[5] = 0x7F800000u;  // +inf
}

__global__ __launch_bounds__(256) void pi_main(const float* __restrict__ SV_A,
                                               const float* __restrict__ SV_N,
                                               const float* __restrict__ MV_A,
                                               const float* __restrict__ MV_N,
                                               float* __restrict__ ws) {
    const int lane = threadIdx.x & 31;
    const int wave = threadIdx.x >> 5;
    const int b    = blockIdx.x * 8 + wave;   // 512 blocks * 8 waves = 4096

    // ---- Phase 1: 12x12 Gram via V_WMMA_F32_16X16X4_F32, K=128 ----
    const int   v      = lane & 15;           // view (A-row M / B-col N)
    const int   kh     = (lane >> 4) << 1;    // K sub-offset: 0 or 2
    const bool  vvalid = (v < N_VIEWS);
    const int   vc     = vvalid ? v : 0;
    const float vmask  = vvalid ? 1.0f : 0.0f;

    const float* aRow = SV_A + ((size_t)vc * BATCH + (size_t)b) * DIM;
    const float* nRow = SV_N + ((size_t)vc * BATCH + (size_t)b) * DIM;

    v8f acc = {};               // D[i][j] = sum_k A[i][k]*N[j][k]
    float pa = 0.0f, pn = 0.0f; // per-lane partial squared norms
#pragma unroll
    for (int k = 0; k < DIM; k += 4) {
        v2f a  = *(const v2f*)(aRow + k + kh);
        v2f nn = *(const v2f*)(nRow + k + kh);
        a  *= vmask;
        nn *= vmask;
        pa += a[0] * a[0] + a[1] * a[1];
        pn += nn[0] * nn[0] + nn[1] * nn[1];
        // (neg_a, A, neg_b, B, c_mod, C, reuse_a, reuse_b)
        acc = __builtin_amdgcn_wmma_f32_16x16x4_f32(false, a, false, nn,
                                                    (short)0, acc, false, false);
    }

    // Combine the two K-halves (lane l and l^16 cover all dims of view l&15)
    float normA = pa + __shfl_xor(pa, 16, 32);
    float normN = pn + __shfl_xor(pn, 16, 32);

    // ---- argmin over 144 pairs. Lane holds j=v, rows i = 8*(lane>>4)+r ----
    const int ibase = (lane >> 4) * 8;
    float bestV = 3.0e38f;
    int   bestI = 0x7FFFFFFF;
#pragma unroll
    for (int r = 0; r < 8; ++r) {
        int   i   = ibase + r;
        float nAi = __shfl(normA, i, 32);      // lanes 0..15 hold normA[view]
        float dv  = nAi + normN - 2.0f * acc[r];
        bool  ok  = (i < N_VIEWS) && vvalid;
        dv = ok ? dv : 3.0e38f;
        int idx = i * N_VIEWS + v;             // flat idx: conf_A=idx/12, conf_N=idx%12
        if (dv < bestV) { bestV = dv; bestI = idx; }
    }
#pragma unroll
    for (int off = 16; off > 0; off >>= 1) {
        float ov = __shfl_xor(bestV, off, 32);
        int   oi = __shfl_xor(bestI, off, 32);
        if (ov < bestV || (ov == bestV && oi < bestI)) { bestV = ov; bestI = oi; }
    }
    const int   confA = bestI / N_VIEWS;
    const int   confN = bestI % N_VIEWS;
    const float mc    = bestV;                 // min squared distance

    // ---- Phase 2: coalesced float4 pass, 4 dims per lane ----
    const float* fa = SV_A + ((size_t)confA * BATCH + (size_t)b) * DIM + lane * 4;
    const float* fn = SV_N + ((size_t)confN * BATCH + (size_t)b) * DIM + lane * 4;
    const float* ma = MV_A + (size_t)b * DIM + lane * 4;
    const float* mn = MV_N + (size_t)b * DIM + lane * 4;

    float4 Fa = *(const float4*)fa;
    float4 Fn = *(const float4*)fn;
    float4 Ma = *(const float4*)ma;
    float4 Mn = *(const float4*)mn;

    float a2 = (Ma.x - Fa.x) * (Ma.x - Fa.x) + (Ma.y - Fa.y) * (Ma.y - Fa.y) +
               (Ma.z - Fa.z) * (Ma.z - Fa.z) + (Ma.w - Fa.w) * (Ma.w - Fa.w);
    float c2 = (Mn.x - Fn.x) * (Mn.x - Fn.x) + (Mn.y - Fn.y) * (Mn.y - Fn.y) +
               (Mn.z - Fn.z) * (Mn.z - Fn.z) + (Mn.w - Fn.w) * (Mn.w - Fn.w);
    float m2 = (Ma.x - Mn.x) * (Ma.x - Mn.x) + (Ma.y - Mn.y) * (Ma.y - Mn.y) +
               (Ma.z - Mn.z) * (Ma.z - Mn.z) + (Ma.w - Mn.w) * (Ma.w - Mn.w);
#pragma unroll
    for (int off = 16; off > 0; off >>= 1) {
        a2 += __shfl_xor(a2, off, 32);
        c2 += __shfl_xor(c2, off, 32);
        m2 += __shfl_xor(m2, off, 32);
    }

    if (lane == 0) {
        float a_cl = sqrtf(a2);
        float c_in = sqrtf(c2);
        float mv   = sqrtf(m2);
        float lm   = fmaxf(a_cl - ALPHA_C, 0.0f);
        float lc   = fmaxf(c_in - ALPHA_C, 0.0f);
        float lmc  = fmaxf(BETA_C - mc, 0.0f);   // uses squared mc, as in reference
        float lmv  = fmaxf(BETA_C - mv, 0.0f);
        float loss = LAMDA_C * (lmc + lmv) + lm + lc;
        float sqmc = sqrtf(mc);

        atomicAdd(&ws[0], loss);
        atomicAdd(&ws[1], (loss != 0.0f) ? 1.0f : 0.0f);
        atomicAdd(&ws[2], sqmc);
        atomicAdd(&ws[3], sqrtf(a_cl));
        atomicAdd(&ws[4], sqrtf(c_in));
        atomicMin((unsigned*)ws + 5, __float_as_uint(sqmc)); // non-negative floats: int order == float order
    }
}

__global__ void pi_final(const float* __restrict__ ws, float* __restrict__ out) {
    if (threadIdx.x == 0) {
        const float inv = 1.0f / (float)BATCH;
        out[0] = ws[0] * inv;                         // avg_loss
        out[1] = ws[1];                               // count_nonzero(losses)
        out[2] = ws[2] * inv;                         // mean sqrt(mc_inter)
        out[3] = 0.5f * (ws[3] + ws[4]) * inv;        // 0.5*(mean sqrt(a)+mean sqrt(c))
        out[4] = __uint_as_float(((const unsigned*)ws)[5]); // min sqrt(mc_inter)
    }
}

extern "C" void kernel_launch(void* const* d_in, const int* in_sizes, int n_in,
                              void* d_out, int out_size, void* d_ws, size_t ws_size,
                              hipStream_t stream) {
    const float* SV_A = (const float*)d_in[0];
    const float* SV_N = (const float*)d_in[1];
    const float* MV_A = (const float*)d_in[2];
    const float* MV_N = (const float*)d_in[3];
    float* ws  = (float*)d_ws;
    float* out = (float*)d_out;

    pi_init<<<1, 64, 0, stream>>>(ws);
    pi_main<<<BATCH / 8, 256, 0, stream>>>(SV_A, SV_N, MV_A, MV_N, ws);
    pi_final<<<1, 64, 0, stream>>>(ws, out);
}